// CTreeOT_87007447482779
// MI455X (gfx1250) — compile-verified
//
#include <hip/hip_runtime.h>
#include <math.h>

// Problem constants (from reference): n0=m0=192, S=384, E=191, 8 steps,
// EPS=0.1, LAM=5  ->  phie = 10*cost^T (padded), psi/EPS = 50*(1-cf).
#define S_DIM 384
#define N0    192
#define EP    192     // edge dimension padded to 192 for 16x16 WMMA tiling
#define E_REAL 191
#define STEPS 8

typedef __attribute__((ext_vector_type(2))) float v2f;
typedef __attribute__((ext_vector_type(8))) float v8f;

// ---------------------------------------------------------------------------
// Setup: build phie, P-transposed matrices (exp space), zero all state.
//   phie[i][j]  = (i<192 && j<192) ? 10*cost[j][i] : 0        (cost_p.T / EPS)
//   Pf[k][i]    = exp(-50*(1-cf[k][i]))  (cf padded per reference)
//   PfT[i][k]   = Pf[k][i]   (A-operand for forward GEMM)
//   PbT[i][k]   = Pb[k][i] = Pf[i][k]  -> PbT is Pf stored row-major
// ---------------------------------------------------------------------------
__global__ void k_setup(const float* __restrict__ cost,
                        const float* __restrict__ constr,
                        float* __restrict__ phie,
                        float* __restrict__ PfT,
                        float* __restrict__ PbT,
                        float* __restrict__ sum_f,
                        float* __restrict__ sum_b,
                        float* __restrict__ msg_f,
                        float* __restrict__ msg_b,
                        float* __restrict__ u,
                        float* __restrict__ v) {
  int idx = blockIdx.x * blockDim.x + threadIdx.x;
  if (idx >= S_DIM * S_DIM) return;
  int i = idx / S_DIM;
  int j = idx - i * S_DIM;

  phie[idx] = (i < N0 && j < N0) ? 10.0f * cost[j * N0 + i] : 0.0f;

  float cf   = (i < N0) ? ((j < N0) ? constr[i * N0 + j] : 0.0f) : 1.0f;
  float pval = expf(-50.0f * (1.0f - cf));   // Pf[i][j]
  PbT[idx]            = pval;                // PbT[i][j] = Pf[i][j]
  PfT[j * S_DIM + i]  = pval;                // PfT[j][i] = Pf[i][j]

  sum_f[idx] = 0.0f;
  sum_b[idx] = 0.0f;
  if (idx < S_DIM * EP) { msg_f[idx] = 0.0f; msg_b[idx] = 0.0f; }
  if (idx < S_DIM)      { u[idx] = 0.0f;     v[idx] = 0.0f;     }
}

// ---------------------------------------------------------------------------
// u[j] = LSE_i( sum_f[i,j]+sum_b[i,j]-phie[i,j] - v[i] )   (online LSE,
// coalesced: thread j walks column j)
// ---------------------------------------------------------------------------
__global__ void k_u(const float* __restrict__ phie,
                    const float* __restrict__ sum_f,
                    const float* __restrict__ sum_b,
                    const float* __restrict__ v,
                    float* __restrict__ u) {
  int j = blockIdx.x * blockDim.x + threadIdx.x;
  if (j >= S_DIM) return;
  float m = -3.4e38f, s = 0.0f;
  for (int i = 0; i < S_DIM; ++i) {
    float x = sum_f[i * S_DIM + j] + sum_b[i * S_DIM + j]
            - phie[i * S_DIM + j] - v[i];
    if (x > m) { s = s * expf(m - x) + 1.0f; m = x; }
    else       { s += expf(x - m); }
  }
  u[j] = m + logf(s);
}

// v[i] = LSE_j( sum_f[i,j]+sum_b[i,j]-phie[i,j] - u[j] )   (uses NEW u)
__global__ void k_v(const float* __restrict__ phie,
                    const float* __restrict__ sum_f,
                    const float* __restrict__ sum_b,
                    const float* __restrict__ u,
                    float* __restrict__ v) {
  int i = blockIdx.x * blockDim.x + threadIdx.x;
  if (i >= S_DIM) return;
  const float* rf = sum_f + i * S_DIM;
  const float* rb = sum_b + i * S_DIM;
  const float* rp = phie  + i * S_DIM;
  float m = -3.4e38f, s = 0.0f;
  for (int j = 0; j < S_DIM; ++j) {
    float x = rf[j] + rb[j] - rp[j] - u[j];
    if (x > m) { s = s * expf(m - x) + 1.0f; m = x; }
    else       { s += expf(x - m); }
  }
  v[i] = m + logf(s);
}

// ---------------------------------------------------------------------------
// Prep: n[e] = max_k(-msg[k,e]);  Q[k,e] = exp(-msg[k,e] - n[e]); pad col=0.
// One block of 192 threads, column-private, coalesced over e.
// ---------------------------------------------------------------------------
__global__ void k_prep(const float* __restrict__ msg_src,
                       float* __restrict__ Q,
                       float* __restrict__ nvec) {
  int e = threadIdx.x;
  float ne = 0.0f;
  if (e < E_REAL) {
    float mn = 3.4e38f;
    for (int k = 0; k < S_DIM; ++k) mn = fminf(mn, msg_src[k * EP + e]);
    ne = -mn;
  }
  nvec[e] = ne;
  for (int k = 0; k < S_DIM; ++k) {
    Q[k * EP + e] = (e < E_REAL) ? expf(-msg_src[k * EP + e] - ne) : 0.0f;
  }
}

// ---------------------------------------------------------------------------
// Core GEMM + fused epilogue.
//   core[i,e] = sum_k AT[i,k] * Q[k,e]           (V_WMMA_F32_16X16X4_F32)
//   lse[i,e]  = n[e] + log(core[i,e])
//   term[i,e] = phie[i,de]+u[de]+v[i]-sum_f[i,de]-sum_b[i,de], de=Edst[e]
//   msg[i,e]  = 0.5*(msg[i,e] + term + lse)
// One 16x16 output tile per wave: 24x12 tiles = 288 waves = 36 blk x 8 waves.
// EXEC is all-ones at every WMMA (no divergence before the epilogue).
// ---------------------------------------------------------------------------
__global__ void k_wmma_msg(const float* __restrict__ AT,     // [S][S], AT[i][k]=P[k][i]
                           const float* __restrict__ Q,      // [S][EP]
                           const float* __restrict__ nvec,   // [EP]
                           const int*   __restrict__ Edge,   // [E][2], dst = Edge[2e+1]
                           const float* __restrict__ phie,
                           const float* __restrict__ u,
                           const float* __restrict__ v,
                           const float* __restrict__ sum_f,
                           const float* __restrict__ sum_b,
                           float* __restrict__ msg) {        // [S][EP] in/out
  int wave = (blockIdx.x * blockDim.x + threadIdx.x) >> 5;   // wave32
  int lane = threadIdx.x & 31;
  if (wave >= (S_DIM / 16) * (EP / 16)) return;              // wave-uniform
  int ti = wave / (EP / 16);
  int tj = wave - ti * (EP / 16);
  int i0 = ti * 16;
  int e0 = tj * 16;

  int m    = lane & 15;          // row (A) / col (B,C) within tile
  int half = lane >> 4;
  int kk   = half << 1;          // K pair owned by this lane half: {0,1} / {2,3}

  v8f c = {0.f, 0.f, 0.f, 0.f, 0.f, 0.f, 0.f, 0.f};
  const float* arow = AT + (size_t)(i0 + m) * S_DIM;
  for (int k0 = 0; k0 < S_DIM; k0 += 4) {
    v2f a, b;
    a.x = arow[k0 + kk];                       // A[m, k0+kk]
    a.y = arow[k0 + kk + 1];                   // A[m, k0+kk+1]
    b.x = Q[(size_t)(k0 + kk)     * EP + e0 + m];   // B[kk,   n=m]
    b.y = Q[(size_t)(k0 + kk + 1) * EP + e0 + m];   // B[kk+1, n=m]
    c = __builtin_amdgcn_wmma_f32_16x16x4_f32(
            /*neg_a=*/false, a, /*neg_b=*/false, b,
            /*c_mod=*/(short)0, c, /*reuse_a=*/false, /*reuse_b=*/false);
  }

  int e = e0 + m;                // this lane's output column
  if (e < E_REAL) {
    int   de  = Edge[2 * e + 1];
    float ne  = nvec[e];
    float ude = u[de];
#pragma unroll
    for (int r = 0; r < 8; ++r) {
      int   i    = i0 + r + 8 * half;          // C/D layout: M = r + 8*(lane/16)
      float lse  = ne + logf(fmaxf(c[r], 1e-38f));
      float term = phie[(size_t)i * S_DIM + de] + ude + v[i]
                 - sum_f[(size_t)i * S_DIM + de]
                 - sum_b[(size_t)i * S_DIM + de];
      float old  = msg[(size_t)i * EP + e];
      msg[(size_t)i * EP + e] = 0.5f * (old + term + lse);
    }
  } else {                                      // padded column e==191
#pragma unroll
    for (int r = 0; r < 8; ++r) {
      int i = i0 + r + 8 * half;
      msg[(size_t)i * EP + e] = 0.0f;
    }
  }
}

// ---------------------------------------------------------------------------
// Scatter: sum[i,s] = sum_{e: dst[e]==s} msg[i,e]   (row-private, no races)
// ---------------------------------------------------------------------------
__global__ void k_scatter(const float* __restrict__ msg,
                          const int*   __restrict__ Edge,
                          float* __restrict__ sum_out) {
  int i = blockIdx.x * blockDim.x + threadIdx.x;
  if (i >= S_DIM) return;
  float* row = sum_out + (size_t)i * S_DIM;
  for (int s = 0; s < S_DIM; ++s) row[s] = 0.0f;
  const float* mrow = msg + (size_t)i * EP;
  for (int e = 0; e < E_REAL; ++e) row[Edge[2 * e + 1]] += mrow[e];
}

// out[j,i] = exp(min(sum_f[i,j]+sum_b[i,j]-phie[i,j]-u[j]-v[i], 0))
__global__ void k_final(const float* __restrict__ phie,
                        const float* __restrict__ sum_f,
                        const float* __restrict__ sum_b,
                        const float* __restrict__ u,
                        const float* __restrict__ v,
                        float* __restrict__ out) {
  int t = blockIdx.x * blockDim.x + threadIdx.x;
  if (t >= S_DIM * S_DIM) return;
  int i = t / S_DIM;
  int j = t - i * S_DIM;
  float lp = sum_f[t] + sum_b[t] - phie[t] - u[j] - v[i];
  out[(size_t)j * S_DIM + i] = expf(fminf(lp, 0.0f));
}

// ---------------------------------------------------------------------------
extern "C" void kernel_launch(void* const* d_in, const int* in_sizes, int n_in,
                              void* d_out, int out_size, void* d_ws, size_t ws_size,
                              hipStream_t stream) {
  const int*   E1f    = (const int*)d_in[0];
  const int*   E1b    = (const int*)d_in[1];
  // d_in[2] = E2f, unused by the forward pass
  const float* cost   = (const float*)d_in[3];
  const float* constr = (const float*)d_in[4];

  const size_t SS = (size_t)S_DIM * S_DIM;   // 147456
  const size_t SE = (size_t)S_DIM * EP;      // 73728
  const size_t need = (5 * SS + 3 * SE + 2 * S_DIM + EP) * sizeof(float);
  if (ws_size < need) return;

  float* p     = (float*)d_ws;
  float* phie  = p; p += SS;
  float* PfT   = p; p += SS;
  float* PbT   = p; p += SS;
  float* sum_f = p; p += SS;
  float* sum_b = p; p += SS;
  float* msg_f = p; p += SE;
  float* msg_b = p; p += SE;
  float* Q     = p; p += SE;
  float* u     = p; p += S_DIM;
  float* v     = p; p += S_DIM;
  float* nvec  = p; p += EP;

  k_setup<<<(SS + 255) / 256, 256, 0, stream>>>(cost, constr, phie, PfT, PbT,
                                                sum_f, sum_b, msg_f, msg_b, u, v);

  for (int step = 0; step < STEPS; ++step) {
    k_u<<<2, 192, 0, stream>>>(phie, sum_f, sum_b, v, u);
    k_v<<<2, 192, 0, stream>>>(phie, sum_f, sum_b, u, v);

    // forward half-step: lse_f from msg_b, update msg_f, rebuild sum_f
    k_prep<<<1, 192, 0, stream>>>(msg_b, Q, nvec);
    k_wmma_msg<<<36, 256, 0, stream>>>(PfT, Q, nvec, E1f, phie, u, v,
                                       sum_f, sum_b, msg_f);
    k_scatter<<<2, 192, 0, stream>>>(msg_f, E1f, sum_f);

    // backward half-step: lse_b from NEW msg_f, uses NEW sum_f (A2)
    k_prep<<<1, 192, 0, stream>>>(msg_f, Q, nvec);
    k_wmma_msg<<<36, 256, 0, stream>>>(PbT, Q, nvec, E1b, phie, u, v,
                                       sum_f, sum_b, msg_b);
    k_scatter<<<2, 192, 0, stream>>>(msg_b, E1b, sum_b);
  }

  k_final<<<(SS + 255) / 256, 256, 0, stream>>>(phie, sum_f, sum_b, u, v,
                                                (float*)d_out);
}